// DenseNGCNLayer_25606595018870
// MI455X (gfx1250) — compile-verified
//
#include <hip/hip_runtime.h>

typedef __attribute__((ext_vector_type(2))) float v2f;
typedef __attribute__((ext_vector_type(8))) float v8f;

#define K_DIM 512
#define N_OUT 128

// ---------------------------------------------------------------------------
// GEMM: X[M,128] = A[M,512] @ W[512,128] using V_WMMA_F32_16X16X4_F32.
//
// Block = 256 threads = 8 waves. Wave w owns output columns [16w, 16w+16).
// Each wave computes FOUR 16x16 M-tiles (64 rows per block), so one B
// fragment feeds four WMMAs (4x less W traffic, higher wmma/vmem density).
//
// Fragment layouts (ISA 7.12.2, wave32):
//   A 16x4  (2 VGPR): lanes 0-15 -> M=lane, K=k0,k0+1; lanes 16-31 -> K=k0+2,k0+3.
//   B 4x16  (2 VGPR): lanes 0-15 -> N=lane, K half 0;  lanes 16-31 -> K half 1.
//   C/D 16x16 (8 VGPR): VGPRi -> M=i (lanes 0-15) / M=i+8 (lanes 16-31), N=lane&15.
// ---------------------------------------------------------------------------
__global__ void __launch_bounds__(256)
gemm_wmma_f32(const float* __restrict__ A, const float* __restrict__ W,
              float* __restrict__ X, int M)
{
    const int wave  = threadIdx.x >> 5;
    const int lane  = threadIdx.x & 31;
    const int lhalf = lane >> 4;     // 0: lanes 0-15, 1: lanes 16-31
    const int l16   = lane & 15;
    const int m0    = blockIdx.x << 6;   // 64 rows per block
    const int n0    = wave << 4;

    const float* __restrict__ Arow0;
    const float* __restrict__ Arow1;
    const float* __restrict__ Arow2;
    const float* __restrict__ Arow3;
    {
        int r0 = m0 +  0 + l16; if (r0 >= M) r0 = M - 1;   // read clamp (tail only)
        int r1 = m0 + 16 + l16; if (r1 >= M) r1 = M - 1;
        int r2 = m0 + 32 + l16; if (r2 >= M) r2 = M - 1;
        int r3 = m0 + 48 + l16; if (r3 >= M) r3 = M - 1;
        Arow0 = A + (size_t)r0 * K_DIM + (lhalf << 1);
        Arow1 = A + (size_t)r1 * K_DIM + (lhalf << 1);
        Arow2 = A + (size_t)r2 * K_DIM + (lhalf << 1);
        Arow3 = A + (size_t)r3 * K_DIM + (lhalf << 1);
    }
    const float* __restrict__ Wcol = W + n0 + l16;

    v8f c0 = {}, c1 = {}, c2 = {}, c3 = {};
#pragma unroll 2
    for (int k0 = 0; k0 < K_DIM; k0 += 4) {
        const int kb = k0 + (lhalf << 1);
        v2f b;
        b.x = Wcol[(size_t)kb * N_OUT];          // W[kb    ][n]
        b.y = Wcol[(size_t)(kb + 1) * N_OUT];    // W[kb + 1][n]
        v2f a0, a1, a2, a3;
        a0.x = Arow0[k0]; a0.y = Arow0[k0 + 1];
        a1.x = Arow1[k0]; a1.y = Arow1[k0 + 1];
        a2.x = Arow2[k0]; a2.y = Arow2[k0 + 1];
        a3.x = Arow3[k0]; a3.y = Arow3[k0 + 1];
        c0 = __builtin_amdgcn_wmma_f32_16x16x4_f32(false, a0, false, b, (short)0, c0, false, false);
        c1 = __builtin_amdgcn_wmma_f32_16x16x4_f32(false, a1, false, b, (short)0, c1, false, false);
        c2 = __builtin_amdgcn_wmma_f32_16x16x4_f32(false, a2, false, b, (short)0, c2, false, false);
        c3 = __builtin_amdgcn_wmma_f32_16x16x4_f32(false, a3, false, b, (short)0, c3, false, false);
    }

    // Epilogue: one uniform bounds decision for the whole block -> the hot
    // path is a branch-free clause of strided b32 stores (no per-row exec
    // masking). Tail block (M not 64-aligned) takes the guarded path.
    float* __restrict__ Xb =
        X + (size_t)(m0 + (lhalf << 3)) * N_OUT + n0 + l16;

    if (m0 + 64 <= M) {
#pragma unroll
        for (int i = 0; i < 8; ++i) Xb[(size_t)(i)      * N_OUT] = c0[i];
#pragma unroll
        for (int i = 0; i < 8; ++i) Xb[(size_t)(i + 16) * N_OUT] = c1[i];
#pragma unroll
        for (int i = 0; i < 8; ++i) Xb[(size_t)(i + 32) * N_OUT] = c2[i];
#pragma unroll
        for (int i = 0; i < 8; ++i) Xb[(size_t)(i + 48) * N_OUT] = c3[i];
    } else {
        const int rbase = m0 + (lhalf << 3);
#pragma unroll
        for (int i = 0; i < 8; ++i) {
            if (rbase + i      < M) Xb[(size_t)(i)      * N_OUT] = c0[i];
            if (rbase + i + 16 < M) Xb[(size_t)(i + 16) * N_OUT] = c1[i];
            if (rbase + i + 32 < M) Xb[(size_t)(i + 32) * N_OUT] = c2[i];
            if (rbase + i + 48 < M) Xb[(size_t)(i + 48) * N_OUT] = c3[i];
        }
    }
}

// ---------------------------------------------------------------------------
// Zero a float buffer (float4 granularity). total4 = element count / 4.
// ---------------------------------------------------------------------------
__global__ void __launch_bounds__(256)
zero4(float4* __restrict__ p, int total4)
{
    const int i = blockIdx.x * 256 + threadIdx.x;
    if (i < total4) p[i] = make_float4(0.f, 0.f, 0.f, 0.f);
}

// ---------------------------------------------------------------------------
// out[n, j] = bias[j]  (accumulator init for the final scatter pass)
// ---------------------------------------------------------------------------
__global__ void __launch_bounds__(256)
bias_init(const float* __restrict__ bias, float* __restrict__ out, int total)
{
    const int i = blockIdx.x * 256 + threadIdx.x;
    if (i < total) out[i] = bias[i & (N_OUT - 1)];
}

// ---------------------------------------------------------------------------
// SpMM scatter: one wave per edge. Lane l moves X[col, 4l..4l+3], scales by
// the edge value, and atomically accumulates into Y[row, 4l..4l+3].
// X (51.2 MB) and Y both live in the 192 MB L2 -> gathers and the
// global_atomic_add_f32 scatters resolve at L2 bandwidth. 128 scalar f32
// atomics per edge is the semantic minimum (no packed f32 atomic exists).
// ---------------------------------------------------------------------------
__global__ void __launch_bounds__(256)
spmm_scatter(const int* __restrict__ rows, const int* __restrict__ cols,
             const float* __restrict__ vals, const float* __restrict__ X,
             float* __restrict__ Y, int E)
{
    const int e = blockIdx.x * 8 + (threadIdx.x >> 5);
    if (e >= E) return;
    const int lane = threadIdx.x & 31;

    const int   r = rows[e];
    const int   c = cols[e];
    const float v = vals[e];

    const float4 x = ((const float4*)(X + (size_t)c * N_OUT))[lane];
    float* dst = Y + (size_t)r * N_OUT + (lane << 2);

    unsafeAtomicAdd(dst + 0, v * x.x);   // global_atomic_add_f32
    unsafeAtomicAdd(dst + 1, v * x.y);
    unsafeAtomicAdd(dst + 2, v * x.z);
    unsafeAtomicAdd(dst + 3, v * x.w);
}

// ---------------------------------------------------------------------------
// Orchestration. Inputs (setup_inputs order):
//   d_in[0] adj_indices [2,E] int      (in_sizes[0] = 2E)
//   d_in[1] adj_values  [E]   f32
//   d_in[2] features    [N,512] f32    (in_sizes[2] = 512N)
//   d_in[3] weight      [512,128] f32
//   d_in[4] bias        [1,128] f32
// d_out: [N,128] f32.  d_ws: two [N,128] f32 ping-pong buffers.
// ---------------------------------------------------------------------------
extern "C" void kernel_launch(void* const* d_in, const int* in_sizes, int n_in,
                              void* d_out, int out_size, void* d_ws, size_t ws_size,
                              hipStream_t stream)
{
    const int*   adj  = (const int*)d_in[0];
    const float* vals = (const float*)d_in[1];
    const float* feat = (const float*)d_in[2];
    const float* wgt  = (const float*)d_in[3];
    const float* bias = (const float*)d_in[4];

    const int E = in_sizes[1];
    const int N = in_sizes[2] / K_DIM;

    const int* rows = adj;          // adj_indices[0, :]
    const int* cols = adj + E;      // adj_indices[1, :]

    float* B0 = (float*)d_ws;
    float* B1 = B0 + (size_t)N * N_OUT;
    float* Dout = (float*)d_out;

    const int total  = N * N_OUT;
    const int total4 = total / 4;
    const dim3 blk(256);
    const dim3 gZero((total4 + 255) / 256);
    const dim3 gBias((total + 255) / 256);
    const dim3 gGemm((N + 63) / 64);
    const dim3 gSpmm((E + 7) / 8);

    // base = features @ weight   -> B0
    gemm_wmma_f32<<<gGemm, blk, 0, stream>>>(feat, wgt, B0, N);

    // iteration 1: B1 = Adj @ B0
    zero4<<<gZero, blk, 0, stream>>>((float4*)B1, total4);
    spmm_scatter<<<gSpmm, blk, 0, stream>>>(rows, cols, vals, B0, B1, E);

    // iteration 2: B0 = Adj @ B1
    zero4<<<gZero, blk, 0, stream>>>((float4*)B0, total4);
    spmm_scatter<<<gSpmm, blk, 0, stream>>>(rows, cols, vals, B1, B0, E);

    // iteration 3 fused with bias: out = bias + Adj @ B0
    bias_init<<<gBias, blk, 0, stream>>>(bias, Dout, total);
    spmm_scatter<<<gSpmm, blk, 0, stream>>>(rows, cols, vals, B0, Dout, E);
}